// Wasserstein_85014582657350
// MI455X (gfx1250) — compile-verified
//
#include <hip/hip_runtime.h>

// Wasserstein (entropic Sinkhorn OT) over 512 graph pairs, NMAX=128, D=64.
// One workgroup per pair; logK resident in LDS for all 500 iterations.
// Cost-matrix cross term computed with V_WMMA_F32_16X16X4_F32.

constexpr int   kNMax  = 128;
constexpr int   kD     = 64;
constexpr int   kS     = 129;     // LDS row stride (odd -> conflict-free rows & cols)
constexpr int   kIters = 500;
constexpr float kEps   = 1e-3f;
constexpr float kNeg   = -1e5f;
constexpr float kNInf  = -3.402823466e38f;

typedef __attribute__((ext_vector_type(2))) float v2f;
typedef __attribute__((ext_vector_type(8))) float v8f;

__global__ __launch_bounds__(128) void wasserstein_sinkhorn_kernel(
    const float* __restrict__ x1, const float* __restrict__ x2,
    const int* __restrict__ sz1, const int* __restrict__ sz2,
    float* __restrict__ out)
{
    __shared__ float sK[kNMax * kS];   // logK = -M/eps   (64.5 KB)
    __shared__ float su[kNMax];
    __shared__ float sv[kNMax];
    __shared__ float sn1[kNMax];       // ||s1_i||^2 ; reused as final reduce buf
    __shared__ float sn2[kNMax];       // ||s2_j||^2

    const int tid = threadIdx.x;
    const int b   = blockIdx.x;

    // ---- exclusive-prefix offsets for this pair (parallel partials + tree) ----
    int* sr1 = (int*)sK;               // alias sK: not live yet
    int* sr2 = (int*)(sK + kNMax);
    int p1 = 0, p2 = 0;
    for (int k = tid; k < b; k += 128) { p1 += sz1[k]; p2 += sz2[k]; }
    sr1[tid] = p1; sr2[tid] = p2;
    __syncthreads();
    for (int off = 64; off > 0; off >>= 1) {
        if (tid < off) { sr1[tid] += sr1[tid + off]; sr2[tid] += sr2[tid + off]; }
        __syncthreads();
    }
    const int off1 = sr1[0];
    const int off2 = sr2[0];
    const int n1   = sz1[b];
    const int n2   = sz2[b];
    // (reads of sr1/sr2 are separated from sK writes by the barrier below)

    // ---- squared norms of packed rows (padded rows are exactly zero) ----
    {
        float a = 0.f, c = 0.f;
        if (tid < n1) {
            const float* r = x1 + (size_t)(off1 + tid) * kD;
            #pragma unroll
            for (int k = 0; k < kD; ++k) a += r[k] * r[k];
        }
        if (tid < n2) {
            const float* r = x2 + (size_t)(off2 + tid) * kD;
            #pragma unroll
            for (int k = 0; k < kD; ++k) c += r[k] * r[k];
        }
        sn1[tid] = a; sn2[tid] = c;
    }
    __syncthreads();

    // ---- cost matrix via V_WMMA_F32_16X16X4_F32: logK[i][j] = -max(n1+n2-2G,0)/eps
    {
        const int wave = tid >> 5;
        const int lane = tid & 31;
        const int h    = lane >> 4;    // lane half: selects K pair (A) / K pair (B)
        const int l16  = lane & 15;    // row (A) / col (B) within 16

        for (int t = wave; t < 64; t += 4) {   // 8x8 tiles of 16x16
            const int tr = t >> 3;
            const int tc = t & 7;
            const int i0 = tr * 16 + l16;      // A row this lane feeds
            const int j0 = tc * 16 + l16;      // B col this lane feeds
            const float am = (i0 < n1) ? 1.f : 0.f;
            const float bm = (j0 < n2) ? 1.f : 0.f;
            const float* ar = x1 + (size_t)(off1 + ((i0 < n1) ? i0 : 0)) * kD + 2 * h;
            const float* br = x2 + (size_t)(off2 + ((j0 < n2) ? j0 : 0)) * kD + 2 * h;

            v8f acc = {};
            #pragma unroll
            for (int k0 = 0; k0 < kD; k0 += 4) {  // 16 WMMA K-steps
                v2f a, bb;
                a[0]  = ar[k0]     * am;   // A: VGPR v holds K = k0 + 2h + v
                a[1]  = ar[k0 + 1] * am;
                bb[0] = br[k0]     * bm;   // B: VGPR v holds K = k0 + 2h + v
                bb[1] = br[k0 + 1] * bm;
                acc = __builtin_amdgcn_wmma_f32_16x16x4_f32(
                    false, a, false, bb, (short)0, acc, false, false);
            }
            #pragma unroll
            for (int r = 0; r < 8; ++r) {  // C: vgpr r -> row tr*16 + r + 8h
                const int i = tr * 16 + r + 8 * h;
                const int j = tc * 16 + l16;
                float m = fmaxf(sn1[i] + sn2[j] - 2.0f * acc[r], 0.0f);
                sK[i * kS + j] = -(m / kEps);
            }
        }
    }

    su[tid] = 0.f;
    sv[tid] = 0.f;
    const float la = (tid < n1) ? 0.f : kNeg;
    const float lb = (tid < n2) ? __logf((float)n1 / (float)n2) : kNeg;
    __syncthreads();

    // ---- 500 Sinkhorn iterations entirely out of LDS ----
    for (int it = 0; it < kIters; ++it) {
        {   // v_j = lb_j - LSE_i(logK[i][j] + u[i])  (thread owns column j)
            const int j = tid;
            float m0 = kNInf, m1 = kNInf, m2 = kNInf, m3 = kNInf;
            for (int i = 0; i < kNMax; i += 4) {
                m0 = fmaxf(m0, sK[(i + 0) * kS + j] + su[i + 0]);
                m1 = fmaxf(m1, sK[(i + 1) * kS + j] + su[i + 1]);
                m2 = fmaxf(m2, sK[(i + 2) * kS + j] + su[i + 2]);
                m3 = fmaxf(m3, sK[(i + 3) * kS + j] + su[i + 3]);
            }
            const float mm = fmaxf(fmaxf(m0, m1), fmaxf(m2, m3));
            float t0 = 0.f, t1 = 0.f, t2 = 0.f, t3 = 0.f;
            for (int i = 0; i < kNMax; i += 4) {
                t0 += __expf(sK[(i + 0) * kS + j] + su[i + 0] - mm);
                t1 += __expf(sK[(i + 1) * kS + j] + su[i + 1] - mm);
                t2 += __expf(sK[(i + 2) * kS + j] + su[i + 2] - mm);
                t3 += __expf(sK[(i + 3) * kS + j] + su[i + 3] - mm);
            }
            sv[j] = lb - (__logf((t0 + t1) + (t2 + t3)) + mm);
        }
        __syncthreads();
        {   // u_i = la_i - LSE_j(logK[i][j] + v[j])  (thread owns row i)
            const int i = tid;
            const float* row = sK + i * kS;
            float m0 = kNInf, m1 = kNInf, m2 = kNInf, m3 = kNInf;
            for (int j = 0; j < kNMax; j += 4) {
                m0 = fmaxf(m0, row[j + 0] + sv[j + 0]);
                m1 = fmaxf(m1, row[j + 1] + sv[j + 1]);
                m2 = fmaxf(m2, row[j + 2] + sv[j + 2]);
                m3 = fmaxf(m3, row[j + 3] + sv[j + 3]);
            }
            const float mm = fmaxf(fmaxf(m0, m1), fmaxf(m2, m3));
            float t0 = 0.f, t1 = 0.f, t2 = 0.f, t3 = 0.f;
            for (int j = 0; j < kNMax; j += 4) {
                t0 += __expf(row[j + 0] + sv[j + 0] - mm);
                t1 += __expf(row[j + 1] + sv[j + 1] - mm);
                t2 += __expf(row[j + 2] + sv[j + 2] - mm);
                t3 += __expf(row[j + 3] + sv[j + 3] - mm);
            }
            su[i] = la - (__logf((t0 + t1) + (t2 + t3)) + mm);
        }
        __syncthreads();
    }

    // ---- out[b] = sum_ij M_ij * P_ij ;  M = -eps*logK, P = exp(logK + u_i + v_j)
    {
        const int j = tid;
        const float vj = sv[j];
        float a0 = 0.f, a1 = 0.f, a2 = 0.f, a3 = 0.f;
        for (int i = 0; i < kNMax; i += 4) {
            float k0 = sK[(i + 0) * kS + j];
            float k1 = sK[(i + 1) * kS + j];
            float k2 = sK[(i + 2) * kS + j];
            float k3 = sK[(i + 3) * kS + j];
            a0 += (-kEps * k0) * __expf(k0 + su[i + 0] + vj);
            a1 += (-kEps * k1) * __expf(k1 + su[i + 1] + vj);
            a2 += (-kEps * k2) * __expf(k2 + su[i + 2] + vj);
            a3 += (-kEps * k3) * __expf(k3 + su[i + 3] + vj);
        }
        sn1[tid] = (a0 + a1) + (a2 + a3);   // sn1 dead since GEMM: reuse
    }
    __syncthreads();
    for (int off = 64; off > 0; off >>= 1) {
        if (tid < off) sn1[tid] += sn1[tid + off];
        __syncthreads();
    }
    if (tid == 0) out[b] = sn1[0];
}

extern "C" void kernel_launch(void* const* d_in, const int* in_sizes, int n_in,
                              void* d_out, int out_size, void* d_ws, size_t ws_size,
                              hipStream_t stream) {
    (void)n_in; (void)d_ws; (void)ws_size; (void)out_size;
    const float* x1  = (const float*)d_in[0];
    const float* x2  = (const float*)d_in[1];
    const int*   sz1 = (const int*)d_in[2];
    const int*   sz2 = (const int*)d_in[3];
    float*       out = (float*)d_out;
    const int bz = in_sizes[2];   // number of graph pairs (512)
    wasserstein_sinkhorn_kernel<<<dim3(bz), dim3(128), 0, stream>>>(
        x1, x2, sz1, sz2, out);
}